// LocalAttention_70480413327459
// MI455X (gfx1250) — compile-verified
//
#include <hip/hip_runtime.h>

typedef __attribute__((ext_vector_type(2))) float v2f;
typedef __attribute__((ext_vector_type(8))) float v8f;

#define NSIDE   64
#define NPATCH  4096
#define NBATCH  8
#define HDIM    32
#define EDIM    34
#define TEDIM   102           // 3*E
#define QST     112           // padded QKV stride (7 n-tiles of 16)
#define TS      8             // interior tile side
#define HS      10            // halo side
#define HTOK    100           // HS*HS
#define MROWS   112           // halo rows padded to 7 m-tiles
#define EK      36            // K padded to 9 steps of 4
#define NTILES  7
#define NJOBS   49            // 7 m-tiles * 7 n-tiles
#define THREADS 128

// LDS layout (float offsets)
#define OFF_EMB   0                        // 112*36  = 4032
#define OFF_W     (OFF_EMB + MROWS*EK)     // 112*36  = 4032
#define OFF_QKV   (OFF_W + QST*EK)         // 100*112 = 11200
#define OFF_CC    (OFF_QKV + HTOK*QST)     // 32*34   = 1088
#define OFF_CB    (OFF_CC + HDIM*EDIM)     // 32
#define OFF_F2    (OFF_CB + HDIM)          // 32*32   = 1024
#define OFF_F2B   (OFF_F2 + HDIM*HDIM)     // 32
#define OFF_LG    (OFF_F2B + HDIM)         // 32
#define OFF_LB    (OFF_LG + HDIM)          // 32
#define LDS_FLOATS (OFF_LB + HDIM)         // 21504 floats = 86016 B

__global__ __launch_bounds__(THREADS) void la_step(
    const float* __restrict__ h_in, size_t in_step,
    float* __restrict__ out, size_t out_step,
    const float* __restrict__ ipw, const float* __restrict__ ipb,
    const float* __restrict__ opw, const float* __restrict__ opb,
    const float* __restrict__ saw, const float* __restrict__ sab,
    const float* __restrict__ f2w, const float* __restrict__ f2b,
    const float* __restrict__ lng, const float* __restrict__ lnb)
{
    extern __shared__ float sm[];
    h_in += (size_t)blockIdx.y * in_step;
    out  += (size_t)blockIdx.y * out_step;

    const int tid = threadIdx.x;
    const int bx  = blockIdx.x;
    const int b   = bx >> 6;
    const int tr  = (bx >> 3) & 7;
    const int tc  = bx & 7;
    const int R0  = tr * TS, C0 = tc * TS;

    float* emb  = sm + OFF_EMB;
    float* wls  = sm + OFF_W;
    float* qkv  = sm + OFF_QKV;
    float* cc   = sm + OFF_CC;
    float* cb   = sm + OFF_CB;
    float* f2   = sm + OFF_F2;
    float* f2bb = sm + OFF_F2B;
    float* lg   = sm + OFF_LG;
    float* lb   = sm + OFF_LB;

    // ---- Phase 0: stage ipw (transposed access, zero-padded), halo embeddings,
    // and the folded post matrices cc = saw@opw, cb = sab + saw@opb ----
    for (int i = tid; i < QST * EK; i += THREADS) {
        int row = i / EK, col = i - row * EK;
        wls[i] = (row < TEDIM && col < EDIM) ? ipw[row * EDIM + col] : 0.f;
    }
    for (int ht = tid; ht < MROWS; ht += THREADS) {
        float* e = emb + ht * EK;                         // 144 B stride -> 16B aligned
        if (ht < HTOK) {
            int hr = ht / HS, hc = ht - hr * HS;
            int gR = R0 - 1 + hr; gR = gR < 0 ? 0 : (gR > 63 ? 63 : gR);
            int gC = C0 - 1 + hc; gC = gC < 0 ? 0 : (gC > 63 ? 63 : gC);
            const float4* src = (const float4*)(h_in +
                (size_t)(b * NPATCH + gR * NSIDE + gC) * HDIM);   // 128B aligned
            float4* ev = (float4*)e;
            #pragma unroll
            for (int k = 0; k < HDIM / 4; ++k) ev[k] = src[k];    // global_load_b128 -> ds b128
            ev[8] = make_float4((float)gR * 0.015625f,            // row/64
                                (float)gC * 0.015625f,            // col/64
                                0.f, 0.f);                        // K padding
        } else {
            float4* ev = (float4*)e;
            #pragma unroll
            for (int k = 0; k < EK / 4; ++k) ev[k] = make_float4(0.f, 0.f, 0.f, 0.f);
        }
    }
    for (int i = tid; i < HDIM * EDIM; i += THREADS) {
        int m = i / EDIM, j = i - m * EDIM;
        float s = 0.f;
        #pragma unroll
        for (int k = 0; k < EDIM; ++k) s += saw[m * EDIM + k] * opw[k * EDIM + j];
        cc[i] = s;
    }
    if (tid < HDIM) {
        float s = sab[tid];
        #pragma unroll
        for (int k = 0; k < EDIM; ++k) s += saw[tid * EDIM + k] * opb[k];
        cb[tid]   = s;
        f2bb[tid] = f2b[tid];
        lg[tid]   = lng[tid];
        lb[tid]   = lnb[tid];
    }
    for (int i = tid; i < (HDIM * HDIM) / 4; i += THREADS)
        ((float4*)f2)[i] = ((const float4*)f2w)[i];
    __syncthreads();

    // ---- Phase 1: QKV[ht, 0:102] = emb @ ipw^T + ipb via V_WMMA_F32_16X16X4_F32 ----
    {
        const int wave  = tid >> 5;
        const int lane  = tid & 31;
        const int lhalf = lane >> 4;      // 0: lanes 0-15, 1: lanes 16-31
        const int l16   = lane & 15;
        for (int job = wave; job < NJOBS; job += THREADS / 32) {   // wave-uniform
            const int mt = job / NTILES, nt = job - mt * NTILES;
            const int am = mt * 16 + l16;                          // A row (M)
            const int bn = nt * 16 + l16;                          // B col (N)
            v8f acc = {0.f, 0.f, 0.f, 0.f, 0.f, 0.f, 0.f, 0.f};
            #pragma unroll
            for (int k0 = 0; k0 < EK; k0 += 4) {
                const int kb = k0 + (lhalf << 1);                  // K{0,1} / K{2,3}
                v2f a;  a.x  = emb[am * EK + kb];  a.y  = emb[am * EK + kb + 1];
                v2f bv; bv.x = wls[bn * EK + kb];  bv.y = wls[bn * EK + kb + 1]; // B[k][n]=ipw[n][k]
                acc = __builtin_amdgcn_wmma_f32_16x16x4_f32(
                    false, a, false, bv, (short)0, acc, false, false);
            }
            const float bias = (bn < TEDIM) ? ipb[bn] : 0.f;
            const int m0 = mt * 16 + (lhalf << 3);
            #pragma unroll
            for (int r = 0; r < 8; ++r) {
                int m = m0 + r;
                if (m < HTOK) qkv[m * QST + bn] = acc[r] + bias;
            }
        }
    }
    __syncthreads();

    // ---- Phase 2+3: per-node center-query attention, folded post-GEMMs, LayerNorm ----
    if (tid < TS * TS) {
        const int ir = tid >> 3, ic = tid & 7;
        const int r = R0 + ir, c = C0 + ic;
        // boundary-corrected neighborhood start (matches nr[0]+=1 / nr[-1]-=1)
        const int sr  = (r == 0) ? 0 : ((r == 63) ? 61 : r - 1);
        const int scc = (c == 0) ? 0 : ((c == 63) ? 61 : c - 1);
        const int hr0 = sr - R0 + 1, hc0 = scc - C0 + 1;

        const float* qp = qkv + ((ir + 1) * HS + (ic + 1)) * QST;  // 448B row stride
        float q[EDIM];
        #pragma unroll
        for (int j = 0; j < EDIM; ++j) q[j] = qp[j];

        float scr[9]; float mx = -1e30f;
        for (int t = 0; t < 9; ++t) {
            const int kht = (hr0 + t / 3) * HS + (hc0 + t % 3);
            const float* kp = qkv + kht * QST + EDIM;
            float d = 0.f;
            #pragma unroll
            for (int j = 0; j < EDIM; ++j) d += q[j] * kp[j];
            d *= 0.17149858514250882f;   // 1/sqrt(34)
            scr[t] = d;
            mx = fmaxf(mx, d);
        }
        float se = 0.f;
        for (int t = 0; t < 9; ++t) { scr[t] = __expf(scr[t] - mx); se += scr[t]; }
        const float sinv = 1.f / se;

        float av[EDIM];
        #pragma unroll
        for (int j = 0; j < EDIM; ++j) av[j] = 0.f;
        for (int t = 0; t < 9; ++t) {
            const int kht = (hr0 + t / 3) * HS + (hc0 + t % 3);
            const float* vp = qkv + kht * QST + 2 * EDIM;
            const float w = scr[t] * sinv;
            #pragma unroll
            for (int j = 0; j < EDIM; ++j) av[j] += w * vp[j];
        }

        const size_t node = (size_t)(b * NPATCH + r * NSIDE + c);
        // residual input: 8x b128 loads
        float hv[HDIM];
        {
            const float4* hp4 = (const float4*)(h_in + node * HDIM);
            #pragma unroll
            for (int k = 0; k < HDIM / 4; ++k) {
                float4 v = hp4[k];
                hv[4 * k + 0] = v.x; hv[4 * k + 1] = v.y;
                hv[4 * k + 2] = v.z; hv[4 * k + 3] = v.w;
            }
        }
        float nh[HDIM];
        #pragma unroll
        for (int m = 0; m < HDIM; ++m) {
            float s = cb[m];
            #pragma unroll
            for (int j = 0; j < EDIM; ++j) s += av[j] * cc[m * EDIM + j];
            nh[m] = hv[m] + s;
        }
        float tm[HDIM]; float mean = 0.f;
        #pragma unroll
        for (int m = 0; m < HDIM; ++m) {
            float s = f2bb[m];
            #pragma unroll
            for (int i = 0; i < HDIM; ++i) s += nh[i] * f2[m * HDIM + i];
            tm[m] = s; mean += s;
        }
        mean *= (1.f / 32.f);
        float var = 0.f;
        #pragma unroll
        for (int m = 0; m < HDIM; ++m) { float d = tm[m] - mean; var += d * d; }
        var *= (1.f / 32.f);
        const float rinv = rsqrtf(var + 1e-5f);
        // output: 8x b128 stores
        float4* op4 = (float4*)(out + node * HDIM);
        #pragma unroll
        for (int k = 0; k < HDIM / 4; ++k) {
            float4 v;
            v.x = (tm[4 * k + 0] - mean) * rinv * lg[4 * k + 0] + lb[4 * k + 0];
            v.y = (tm[4 * k + 1] - mean) * rinv * lg[4 * k + 1] + lb[4 * k + 1];
            v.z = (tm[4 * k + 2] - mean) * rinv * lg[4 * k + 2] + lb[4 * k + 2];
            v.w = (tm[4 * k + 3] - mean) * rinv * lg[4 * k + 3] + lb[4 * k + 3];
            op4[k] = v;
        }
    }
}

extern "C" void kernel_launch(void* const* d_in, const int* in_sizes, int n_in,
                              void* d_out, int out_size, void* d_ws, size_t ws_size,
                              hipStream_t stream) {
    (void)in_sizes; (void)n_in; (void)out_size; (void)d_ws; (void)ws_size;
    const float* x   = (const float*)d_in[0];
    const float* ipw = (const float*)d_in[1];
    const float* ipb = (const float*)d_in[2];
    const float* opw = (const float*)d_in[3];
    const float* opb = (const float*)d_in[4];
    const float* saw = (const float*)d_in[5];
    const float* sab = (const float*)d_in[6];
    const float* f2w = (const float*)d_in[7];
    const float* f2b = (const float*)d_in[8];
    const float* lng = (const float*)d_in[9];
    const float* lnb = (const float*)d_in[10];
    float* out = (float*)d_out;

    const size_t step  = (size_t)NBATCH * NPATCH * HDIM;  // 1,048,576 floats
    const size_t shmem = LDS_FLOATS * sizeof(float);      // 86,016 B (<< 320 KB/WGP)
    static bool attr_set = false;  // runtime-config only; not a stream op, no output state
    if (!attr_set) {
        hipFuncSetAttribute((const void*)la_step,
                            hipFuncAttributeMaxDynamicSharedMemorySize, (int)shmem);
        attr_set = true;
    }

    // 10 independent observation steps in one launch (gridDim.y = time)
    la_step<<<dim3(512, 10), THREADS, shmem, stream>>>(
        x, step, out, step, ipw, ipb, opw, opb, saw, sab, f2w, f2b, lng, lnb);
    // 50 strictly sequential prediction steps, chained on the stream
    for (int t = 10; t < 60; ++t) {
        la_step<<<dim3(512, 1), THREADS, shmem, stream>>>(
            out + (size_t)(t - 1) * step, 0, out + (size_t)t * step, 0,
            ipw, ipb, opw, opb, saw, sab, f2w, f2b, lng, lnb);
    }
}